// SAGEConvModel_7361573945898
// MI455X (gfx1250) — compile-verified
//
#include <hip/hip_runtime.h>

#define NN 100000
#define NE 600000
#define CH 128

typedef __attribute__((ext_vector_type(2))) float v2f;
typedef __attribute__((ext_vector_type(8))) float v8f;

// ---------------- degree ----------------
__global__ __launch_bounds__(256) void k_deg(const int* __restrict__ dst,
                                             float* __restrict__ deg) {
  int e = blockIdx.x * blockDim.x + threadIdx.x;
  if (e < NE) unsafeAtomicAdd(&deg[dst[e]], 1.0f);
}

__global__ __launch_bounds__(256) void k_deginv(float* __restrict__ deg) {
  int n = blockIdx.x * blockDim.x + threadIdx.x;
  if (n < NN) {
    float d = deg[n];
    deg[n] = (d > 0.0f) ? (1.0f / d) : 0.0f;
  }
}

// ---------------- layer-1 scatter (3 features) ----------------
__global__ __launch_bounds__(256) void k_scatter3(const float* __restrict__ x,
                                                  const int* __restrict__ src,
                                                  const int* __restrict__ dst,
                                                  float* __restrict__ agg3) {
  int e = blockIdx.x * blockDim.x + threadIdx.x;
  if (e >= NE) return;
  int s = src[e], d = dst[e];
  const float* p = x + (size_t)s * 3;
  float* q = agg3 + (size_t)d * 3;
  unsafeAtomicAdd(q + 0, p[0]);
  unsafeAtomicAdd(q + 1, p[1]);
  unsafeAtomicAdd(q + 2, p[2]);
}

// ---------------- layer-1 dense (K=3, scalar VALU) ----------------
__global__ __launch_bounds__(256) void k_layer1(const float* __restrict__ x,
                                                const float* __restrict__ agg3,
                                                const float* __restrict__ deginv,
                                                const float* __restrict__ W1l,
                                                const float* __restrict__ W1r,
                                                const float* __restrict__ b1,
                                                float* __restrict__ h1) {
  int tid = blockIdx.x * blockDim.x + threadIdx.x;
  if (tid >= NN * CH) return;
  int n = tid >> 7;
  int c = tid & (CH - 1);
  float dinv = deginv[n];
  float s = b1[c];
#pragma unroll
  for (int k = 0; k < 3; ++k) {
    s += agg3[n * 3 + k] * dinv * W1l[k * CH + c];
    s += x[n * 3 + k] * W1r[k * CH + c];
  }
  h1[tid] = fmaxf(s, 0.0f);
}

// ---------------- scatter for 128-wide features ----------------
// one thread handles (edge, 4 channels) via float4 gather + 4 hw fp atomics
__global__ __launch_bounds__(256) void k_scatter128(const float* __restrict__ hin,
                                                    const int* __restrict__ src,
                                                    const int* __restrict__ dst,
                                                    float* __restrict__ agg) {
  int tid = blockIdx.x * blockDim.x + threadIdx.x;
  if (tid >= NE * 32) return;
  int e = tid >> 5;
  int c = (tid & 31) << 2;
  int s = src[e], d = dst[e];
  float4 v = *reinterpret_cast<const float4*>(hin + (size_t)s * CH + c);
  float* q = agg + (size_t)d * CH + c;
  unsafeAtomicAdd(q + 0, v.x);
  unsafeAtomicAdd(q + 1, v.y);
  unsafeAtomicAdd(q + 2, v.z);
  unsafeAtomicAdd(q + 3, v.w);
}

// ---------------- fused SAGE dense layer via fp32 WMMA ----------------
// out[n, :] = relu( (agg[n,:]*deginv[n]) @ Wl + hin[n,:] @ Wr + b )
// One wave computes one 16x16 output tile; 4 waves/block share the 16 rows.
__global__ __launch_bounds__(128) void k_sage_gemm_wmma(
    const float* __restrict__ hin,    // [NN, CH]
    const float* __restrict__ agg,    // [NN, CH]  (unscaled sums)
    const float* __restrict__ deginv, // [NN]
    const float* __restrict__ Wl,     // [CH, CH]
    const float* __restrict__ Wr,     // [CH, CH]
    const float* __restrict__ bias,   // [CH]
    float* __restrict__ hout)         // [NN, CH]
{
  const int lane = threadIdx.x & 31;
  const int wave = threadIdx.x >> 5;
  const int rowTile = blockIdx.x;               // 0..6249  (NN/16)
  const int colTile = blockIdx.y * 4 + wave;    // 0..7     (CH/16)

  const int m = lane & 15;                      // A-row / B-col within tile
  const int kb = (lane >> 4) << 1;              // K-pair base: 0 or 2
  const int row = rowTile * 16 + m;
  const int col = colTile * 16 + m;

  const float dinv = deginv[row];
  const float* aRow = agg + (size_t)row * CH;
  const float* hRow = hin + (size_t)row * CH;

  v8f acc = {};
#pragma unroll 4
  for (int k = 0; k < CH; k += 4) {
    v2f a, b;
    // agg (scaled) x Wl
    a.x = aRow[k + kb] * dinv;
    a.y = aRow[k + kb + 1] * dinv;
    b.x = Wl[(k + kb) * CH + col];
    b.y = Wl[(k + kb + 1) * CH + col];
    acc = __builtin_amdgcn_wmma_f32_16x16x4_f32(false, a, false, b,
                                                (short)0, acc, false, false);
    // hin x Wr
    a.x = hRow[k + kb];
    a.y = hRow[k + kb + 1];
    b.x = Wr[(k + kb) * CH + col];
    b.y = Wr[(k + kb + 1) * CH + col];
    acc = __builtin_amdgcn_wmma_f32_16x16x4_f32(false, a, false, b,
                                                (short)0, acc, false, false);
  }

  // C/D layout: VGPR r -> M = r (lanes 0-15) or r+8 (lanes 16-31), N = lane&15
  const int mBase = rowTile * 16 + ((lane >> 4) << 3);
  const float bc = bias[col];
#pragma unroll
  for (int r = 0; r < 8; ++r) {
    float v = acc[r] + bc;
    hout[(size_t)(mBase + r) * CH + col] = fmaxf(v, 0.0f);
  }
}

extern "C" void kernel_launch(void* const* d_in, const int* in_sizes, int n_in,
                              void* d_out, int out_size, void* d_ws, size_t ws_size,
                              hipStream_t stream) {
  const float* x   = (const float*)d_in[0];
  const int*   ei  = (const int*)d_in[1];
  const int*   src = ei;            // edge_index[0]
  const int*   dst = ei + NE;       // edge_index[1]
  const float* W1l = (const float*)d_in[2];
  const float* W1r = (const float*)d_in[3];
  const float* b1  = (const float*)d_in[4];
  const float* W2l = (const float*)d_in[5];
  const float* W2r = (const float*)d_in[6];
  const float* b2  = (const float*)d_in[7];
  const float* W3l = (const float*)d_in[8];
  const float* W3r = (const float*)d_in[9];
  const float* b3  = (const float*)d_in[10];
  float* out = (float*)d_out;
  float* ws  = (float*)d_ws;

  // workspace layout (floats, 64-float aligned)
  const size_t OFF_DEG  = 0;                       // NN
  const size_t OFF_AGG3 = 100096;                  // NN*3
  const size_t OFF_A    = 400128;                  // NN*CH  (h1)
  const size_t OFF_B    = OFF_A + (size_t)NN * CH; // NN*CH  (agg scratch)
  const size_t OFF_C    = OFF_B + (size_t)NN * CH; // NN*CH  (h2)
  float* deg  = ws + OFF_DEG;
  float* agg3 = ws + OFF_AGG3;
  float* h1   = ws + OFF_A;
  float* aggB = ws + OFF_B;
  float* h2   = ws + OFF_C;

  const int T = 256;
  const dim3 gemmGrid(NN / 16, 2);
  const dim3 gemmBlock(128);

  // ---- degree + deg_inv ----
  hipMemsetAsync(deg, 0, sizeof(float) * NN, stream);
  hipMemsetAsync(agg3, 0, sizeof(float) * NN * 3, stream);
  k_deg<<<(NE + T - 1) / T, T, 0, stream>>>(dst, deg);
  k_deginv<<<(NN + T - 1) / T, T, 0, stream>>>(deg);

  // ---- layer 1 ----
  k_scatter3<<<(NE + T - 1) / T, T, 0, stream>>>(x, src, dst, agg3);
  k_layer1<<<(NN * CH + T - 1) / T, T, 0, stream>>>(x, agg3, deg, W1l, W1r, b1, h1);

  // ---- layer 2 ----
  hipMemsetAsync(aggB, 0, sizeof(float) * (size_t)NN * CH, stream);
  k_scatter128<<<(NE * 32 + T - 1) / T, T, 0, stream>>>(h1, src, dst, aggB);
  k_sage_gemm_wmma<<<gemmGrid, gemmBlock, 0, stream>>>(h1, aggB, deg, W2l, W2r, b2, h2);

  // ---- layer 3 ----
  hipMemsetAsync(aggB, 0, sizeof(float) * (size_t)NN * CH, stream);
  k_scatter128<<<(NE * 32 + T - 1) / T, T, 0, stream>>>(h2, src, dst, aggB);
  k_sage_gemm_wmma<<<gemmGrid, gemmBlock, 0, stream>>>(h2, aggB, deg, W3l, W3r, b3, out);
}